// GraphConvolution_4595615007340
// MI455X (gfx1250) — compile-verified
//
#include <hip/hip_runtime.h>
#include <hip/hip_bf16.h>

#define IN_F   256
#define OUT_F  128
#define SH_STRIDE 144   // LDS row stride in floats: 128 + 16 pad (bank-conflict free)

typedef float v2f __attribute__((ext_vector_type(2)));
typedef float v8f __attribute__((ext_vector_type(8)));
typedef unsigned int u32x4 __attribute__((ext_vector_type(4)));
typedef int i32x4 __attribute__((ext_vector_type(4)));
typedef int i32x8 __attribute__((ext_vector_type(8)));

// ---------------------------------------------------------------------------
// Kernel 1: support = X @ W via V_WMMA_F32_16X16X4_F32 (fp32 WMMA, wave32).
// W (256x128 f32, 128 KB) is DMA'd once per workgroup into LDS by the Tensor
// Data Mover (2D D#, data_size=4B, padded rows: 16 DWORDs pad every 128 DWORDs
// -> LDS stride 144 floats so the two half-wave K-rows hit disjoint banks).
// One wave computes a 16x128 row-strip of `support` (8 16x16 accumulators);
// A fragments stream from global (b64), B fragments come from LDS (ds_load).
// Fragment layouts per ISA 7.12.2 (A 16x4 f32: lanes 0-15 K={0,1},
// lanes 16-31 K={2,3}; C/D: VGPR v -> M=v(+8 for upper half), N=lane%16).
// ---------------------------------------------------------------------------
__global__ __launch_bounds__(256) void gcn_gemm_wmma(
    const float* __restrict__ X, const float* __restrict__ W,
    float* __restrict__ S, int mTiles) {
  extern __shared__ float shW[];                   // IN_F * SH_STRIDE floats
  const int lane = threadIdx.x & 31;
  const int wave = threadIdx.x >> 5;

  if (wave == 0) {
    // Build Tensor DMA Descriptor (ISA 8.3/8.4) and issue TENSOR_LOAD_TO_LDS.
    const unsigned long long ga = (unsigned long long)(uintptr_t)W;
    const unsigned int lds = (unsigned int)(uintptr_t)shW;  // addr[31:0] == LDS offset
    u32x4 g0;
    g0.x = 1u;                                     // count=1 (valid), no gather
    g0.y = lds;                                    // lds_addr [63:32]
    g0.z = (unsigned int)(ga & 0xFFFFFFFFu);       // global_addr[31:0]
    g0.w = (unsigned int)((ga >> 32) & 0x01FFFFFFu)// global_addr[56:32]
         | (2u << 30);                             // type=2 ("image")
    i32x8 g1;
    g1[0] = (2 << 16)                              // data_size=2 -> 4 bytes
          | (1 << 20)                              // pad_enable (LDS dest pad)
          | (6 << 22)                              // pad_interval: every 128 DWORDs
          | (15 << 25);                            // pad_amount: 16 DWORDs
    g1[1] = (OUT_F & 0xFFFF) << 16;                // tensor_dim0 = 128 (lo16)
    g1[2] = (IN_F & 0xFFFF) << 16;                 // dim0 hi=0 | tensor_dim1 = 256
    g1[3] = (OUT_F << 16);                         // dim1 hi=0 | tile_dim0 = 128
    g1[4] = IN_F;                                  // tile_dim1 = 256, tile_dim2 = 0
    g1[5] = OUT_F;                                 // tensor_dim0_stride = 128 elems
    g1[6] = 0;
    g1[7] = 0;
    const i32x4 z4 = {0, 0, 0, 0};                 // groups 2/3 unused (2D tensor)
    const i32x8 z8 = {0, 0, 0, 0, 0, 0, 0, 0};
    __builtin_amdgcn_tensor_load_to_lds(g0, g1, z4, z4, z8, 0);
    __builtin_amdgcn_s_wait_tensorcnt(0);          // data resident before barrier
  }
  __syncthreads();                                 // all waves see staged W

  const int tile = blockIdx.x * 8 + wave;          // one 16-row strip per wave
  if (tile >= mTiles) return;                      // wave-uniform: EXEC stays all-1s

  const int l15  = lane & 15;
  const int kOff = (lane >> 4) << 1;               // 0 (lanes 0-15) / 2 (lanes 16-31)

  const float* Xp = X + (size_t)(tile * 16 + l15) * IN_F + kOff;   // A: M=l15
  const int wBase = kOff * SH_STRIDE + l15;                        // B: N=l15

  v8f acc[8];
#pragma unroll
  for (int n = 0; n < 8; ++n) acc[n] = (v8f){0.f,0.f,0.f,0.f,0.f,0.f,0.f,0.f};

#pragma unroll 4
  for (int k = 0; k < IN_F; k += 4) {
    v2f a;
    a.x = Xp[k];
    a.y = Xp[k + 1];
    const int sBase = wBase + k * SH_STRIDE;
#pragma unroll
    for (int n = 0; n < 8; ++n) {
      v2f b;
      b.x = shW[sBase + n * 16];                   // W[k+kOff,   n*16+l15]
      b.y = shW[sBase + SH_STRIDE + n * 16];       // W[k+kOff+1, n*16+l15]
      acc[n] = __builtin_amdgcn_wmma_f32_16x16x4_f32(
          /*neg_a=*/false, a, /*neg_b=*/false, b,
          /*c_mod=*/(short)0, acc[n], /*reuse_a=*/false, /*reuse_b=*/false);
    }
  }

  // Store D: VGPR v holds row M = v (+8 for upper half-wave), col = l15.
  const int mBase = tile * 16 + ((lane >> 4) << 3);
  float* Sp = S + (size_t)mBase * OUT_F + l15;
#pragma unroll
  for (int n = 0; n < 8; ++n) {
#pragma unroll
    for (int v = 0; v < 8; ++v) {
      Sp[(size_t)v * OUT_F + n * 16] = acc[n][v];
    }
  }
}

// ---------------------------------------------------------------------------
// Kernel 2: out[i, :] = b  (float4 broadcast init; must precede the scatter)
// ---------------------------------------------------------------------------
__global__ __launch_bounds__(256) void gcn_bias_init(
    const float* __restrict__ b, float* __restrict__ out, int total4) {
  int idx = blockIdx.x * 256 + threadIdx.x;        // over N*OUT_F/4 float4s
  if (idx >= total4) return;
  int f4 = idx & (OUT_F / 4 - 1);                  // 32 float4 per row
  float4 bv = reinterpret_cast<const float4*>(b)[f4];
  reinterpret_cast<float4*>(out)[idx] = bv;
}

// ---------------------------------------------------------------------------
// Kernel 3: COO SpMM scatter. One wave per edge; lane owns 4 contiguous
// floats of the 128-wide row. `support` (51.2 MB) is L2-resident (192 MB L2),
// so both the gather and the hardware f32 atomics stay on-chip.
// Edge metadata is wave-uniform -> scalar loads.
// ---------------------------------------------------------------------------
__global__ __launch_bounds__(256) void gcn_spmm_scatter(
    const int* __restrict__ rows, const int* __restrict__ cols,
    const float* __restrict__ vals, const float* __restrict__ S,
    float* __restrict__ out, int nEdges) {
  const int lane = threadIdx.x & 31;
  const int e = (int)((blockIdx.x * 256 + threadIdx.x) >> 5);
  if (e >= nEdges) return;

  const int   r = rows[e];
  const int   c = cols[e];
  const float v = vals[e];

  const float4 s = *reinterpret_cast<const float4*>(S + (size_t)c * OUT_F + lane * 4);
  float* o = out + (size_t)r * OUT_F + lane * 4;
  unsafeAtomicAdd(o + 0, v * s.x);   // global_atomic_add_f32, non-returning
  unsafeAtomicAdd(o + 1, v * s.y);
  unsafeAtomicAdd(o + 2, v * s.z);
  unsafeAtomicAdd(o + 3, v * s.w);
}

// ---------------------------------------------------------------------------
extern "C" void kernel_launch(void* const* d_in, const int* in_sizes, int n_in,
                              void* d_out, int out_size, void* d_ws, size_t ws_size,
                              hipStream_t stream) {
  const float* x    = (const float*)d_in[0];
  const int*   erow = (const int*)  d_in[1];
  const int*   ecol = (const int*)  d_in[2];
  const float* eval = (const float*)d_in[3];
  const float* w    = (const float*)d_in[4];
  const float* bias = (const float*)d_in[5];
  float*       out  = (float*)d_out;
  float*       supp = (float*)d_ws;                // N*OUT_F*4 = 51.2 MB scratch

  const int nNodes = in_sizes[0] / IN_F;           // 100000
  const int nEdges = in_sizes[1];                  // 1600000

  // 1) support = X @ W  (fp32 WMMA + TDM-staged W in LDS). 100000 = 6250*16.
  const int mTiles = nNodes / 16;
  const size_t ldsBytes = (size_t)IN_F * SH_STRIDE * sizeof(float);  // 144 KB
  gcn_gemm_wmma<<<(mTiles + 7) / 8, 256, ldsBytes, stream>>>(x, w, supp, mTiles);

  // 2) out = broadcast(b)
  const int total4 = nNodes * (OUT_F / 4);
  gcn_bias_init<<<(total4 + 255) / 256, 256, 0, stream>>>(bias, out, total4);

  // 3) out += A_coo @ support  (atomic scatter, one wave per edge)
  const long long threads = (long long)nEdges * 32;
  gcn_spmm_scatter<<<(int)((threads + 255) / 256), 256, 0, stream>>>(
      erow, ecol, eval, supp, out, nEdges);
}